// CNOTRing_18811956756652
// MI455X (gfx1250) — compile-verified
//
#include <hip/hip_runtime.h>
#include <stdint.h>

// CNOT ring on 20 wires == GF(2)-linear index permutation:
//   src(k) = (k ^ (k>>1)) ^ ((k&1) ? 0xC0000 : 0)
// Pure bandwidth problem (512 MB traffic, ~22us at 23.3 TB/s).
// Strategy: LDS-staged block permutation so every HBM access is a full
// 128B transaction, using the CDNA5 async global->LDS path (ASYNCcnt).

#define LDS_ELEMS 4096u   // 2^12 float2 per staging buffer (32 KB each)

typedef int   v4i __attribute__((ext_vector_type(4)));
typedef float v4f __attribute__((ext_vector_type(4)));
typedef float v2f __attribute__((ext_vector_type(2)));

typedef __attribute__((address_space(1))) v4i glb_v4i;
typedef __attribute__((address_space(3))) v4i lds_v4i;

__device__ __forceinline__ void async_load_b128(const void* gsrc, void* ldst) {
#if __has_builtin(__builtin_amdgcn_global_load_async_to_lds_b128)
    __builtin_amdgcn_global_load_async_to_lds_b128(
        (glb_v4i*)gsrc, (lds_v4i*)ldst, /*offset=*/0, /*cpol=*/0);
#else
    unsigned lofs = (unsigned)(uintptr_t)(lds_v4i*)ldst;
    unsigned long long ga = (unsigned long long)(uintptr_t)gsrc;
    asm volatile("global_load_async_to_lds_b128 %0, %1, off"
                 :: "v"(lofs), "v"(ga) : "memory");
#endif
}

__device__ __forceinline__ void wait_async0() {
#if __has_builtin(__builtin_amdgcn_s_wait_asynccnt)
    __builtin_amdgcn_s_wait_asynccnt(0);
#else
    asm volatile("s_wait_asynccnt 0" ::: "memory");
#endif
}

__global__ __launch_bounds__(256)
void CNOTRing_kernel(const v2f* __restrict__ in, v2f* __restrict__ out) {
    __shared__ v2f buf0[LDS_ELEMS];   // source block GH
    __shared__ v2f buf1[LDS_ELEMS];   // source block GH ^ 0xC0000

    const unsigned tid   = threadIdx.x;
    const unsigned w     = blockIdx.x;          // 4096 workgroups
    const unsigned batch = w >> 7;              // 32 batches
    const unsigned blk   = w & 127u;            // 128 block-pairs per batch
    const unsigned K     = blk << 12;           // output block A (bit19 = 0)
    const unsigned K2    = K | 0x80000u;        // partner output block (bit19 = 1)
    const unsigned GH    = (K ^ (K >> 1)) & 0xFF000u;  // gray(K), high bits
    const unsigned GH2   = GH ^ 0xC0000u;
    const unsigned Kb    = (K >> 12) & 1u;      // carry-in bit for low-gray map
    const size_t   base  = (size_t)batch << 20; // float2 elements per batch

    const v2f* __restrict__ src0 = in + base + GH;
    const v2f* __restrict__ src1 = in + base + GH2;

    // Stage 2 x 32 KB contiguous source blocks into LDS: 16B per lane per op,
    // 8 rounds per buffer -> fully coalesced async b128 loads.
    #pragma unroll
    for (unsigned r = 0; r < 8; ++r) {
        unsigned e = (r * 256u + tid) * 2u;     // float2 element index (16B granule)
        async_load_b128(src0 + e, &buf0[e]);
        async_load_b128(src1 + e, &buf1[e]);
    }
    wait_async0();
    __syncthreads();

    v2f* __restrict__ dstA = out + base + K;
    v2f* __restrict__ dstB = out + base + K2;

    // out[K  + kl] = (kl even ? buf0 : buf1)[glow(kl)]
    // out[K2 + kl] = (kl even ? buf1 : buf0)[glow(kl)]
    // with glow(kl) = (kl ^ (kl>>1)) ^ (Kb<<11);  glow(kl+1) = glow(kl)^1 for even kl.
    #pragma unroll
    for (unsigned s = 0; s < 8; ++s) {
        unsigned kl = 2u * (s * 256u + tid);                     // even
        unsigned g  = ((kl ^ (kl >> 1)) ^ (Kb << 11)) & 0xFFFu;
        unsigned gp = g ^ 1u;
        v2f a0 = buf0[g];   // -> dstA[kl]
        v2f a1 = buf1[gp];  // -> dstA[kl+1]
        v2f b0 = buf1[g];   // -> dstB[kl]
        v2f b1 = buf0[gp];  // -> dstB[kl+1]
        v4f va = {a0.x, a0.y, a1.x, a1.y};
        v4f vb = {b0.x, b0.y, b1.x, b1.y};
        __builtin_nontemporal_store(va, (v4f*)(dstA + kl));      // 16B coalesced NT
        __builtin_nontemporal_store(vb, (v4f*)(dstB + kl));
    }
}

extern "C" void kernel_launch(void* const* d_in, const int* in_sizes, int n_in,
                              void* d_out, int out_size, void* d_ws, size_t ws_size,
                              hipStream_t stream) {
    const v2f* in  = (const v2f*)d_in[0];   // [32][2^20] float2 (re,im)
    v2f*       out = (v2f*)d_out;
    dim3 grid(32u * 128u);   // batch * block-pairs
    dim3 block(256u);
    hipLaunchKernelGGL(CNOTRing_kernel, grid, block, 0, stream, in, out);
}